// Sampler_69887707841060
// MI455X (gfx1250) — compile-verified
//
#include <hip/hip_runtime.h>
#include <math.h>
#include <stdint.h>

#define BATCH    128
#define VOCAB    128000
#define SPLIT    5                       // segments per row for streaming passes
#define SEG      (VOCAB / SPLIT)         // 25600 floats
#define A_THREADS 320                    // 10 wave32s
#define A_WAVES  (A_THREADS / 32)
#define A_TILE   (A_THREADS * 8)         // 2560 floats = 10 KB per buffer
#define A_NTILES (SEG / A_TILE)          // 10, exact
#define NBUF     3                       // triple buffer, 2 tiles in flight
#define D_THREADS 640                    // 20 wave32s for selection kernel
#define D_WAVES  (D_THREADS / 32)
#define CHUNK    (VOCAB / D_THREADS)     // 200, exact
#define KMAX     64
#define F32_LOWEST (-3.402823466e38f)
#define STATS_OFF (BATCH * SPLIT * 2)    // float offset of per-row stats in ws

__device__ __forceinline__ float uniform_hash(uint32_t x) {
    x ^= x >> 16; x *= 0x7feb352dU;
    x ^= x >> 15; x *= 0x846ca68bU;
    x ^= x >> 16;
    return ((float)x + 0.5f) * (1.0f / 4294967296.0f);
}

__device__ __forceinline__ void async_load_b128(uint32_t lds_off, const float* gaddr) {
    // CDNA5 async copy global -> LDS, tracked by ASYNCcnt (per-lane LDS offset in %0)
    asm volatile("global_load_async_to_lds_b128 %0, %1, off"
                 :: "v"(lds_off), "v"(gaddr) : "memory");
}

// ---------------- Kernel A: per-segment online softmax (the one HBM read) ----------------
__global__ __launch_bounds__(A_THREADS)
void partial_softmax_kernel(const float* __restrict__ logits,
                            const float* __restrict__ temps,
                            float* __restrict__ ws)
{
    __shared__ float buf[NBUF][A_TILE];
    __shared__ float redM[A_WAVES];
    __shared__ float redS[A_WAVES];

    const int row  = blockIdx.x / SPLIT;
    const int seg  = blockIdx.x % SPLIT;
    const int tid  = threadIdx.x;
    const int lane = tid & 31;
    const int wave = tid >> 5;
    const float invT = 1.0f / temps[row];
    const float* segp = logits + (size_t)row * VOCAB + (size_t)seg * SEG;

    // prologue: two tiles in flight (4 outstanding b128 asyncs per lane)
    #pragma unroll
    for (int p = 0; p < 2; ++p) {
        uint32_t l0 = (uint32_t)(uintptr_t)(&buf[p][tid * 8]);
        const float* g = segp + (size_t)p * A_TILE + tid * 8;
        async_load_b128(l0, g);
        async_load_b128(l0 + 16, g + 4);
    }

    float m = -INFINITY, s = 0.0f;
    for (int t = 0; t < A_NTILES; ++t) {
        if (t + 2 < A_NTILES) {
            int p = (t + 2) % NBUF;
            uint32_t l0 = (uint32_t)(uintptr_t)(&buf[p][tid * 8]);
            const float* g = segp + (size_t)(t + 2) * A_TILE + tid * 8;
            async_load_b128(l0, g);
            async_load_b128(l0 + 16, g + 4);
            asm volatile("s_wait_asynccnt 0x4" ::: "memory");   // tile t landed
        } else if (t + 1 < A_NTILES) {
            asm volatile("s_wait_asynccnt 0x2" ::: "memory");
        } else {
            asm volatile("s_wait_asynccnt 0x0" ::: "memory");
        }
        const float* cur = buf[t % NBUF];
        float4 a = *reinterpret_cast<const float4*>(&cur[tid * 8]);
        float4 b = *reinterpret_cast<const float4*>(&cur[tid * 8 + 4]);
        float xs[8] = { a.x, a.y, a.z, a.w, b.x, b.y, b.z, b.w };
        #pragma unroll
        for (int j = 0; j < 8; ++j) {
            float x  = xs[j] * invT;
            float nm = fmaxf(m, x);
            s = s * __expf(m - nm) + __expf(x - nm);
            m = nm;
        }
    }
    // wave32 (m,s) pair-combine, then cross-wave
    #pragma unroll
    for (int off = 16; off > 0; off >>= 1) {
        float om = __shfl_xor(m, off, 32);
        float os = __shfl_xor(s, off, 32);
        float nm = fmaxf(m, om);
        s = s * __expf(m - nm) + os * __expf(om - nm);
        m = nm;
    }
    if (lane == 0) { redM[wave] = m; redS[wave] = s; }
    __syncthreads();
    if (wave == 0) {
        float mm = (lane < A_WAVES) ? redM[lane] : -INFINITY;
        float ss = (lane < A_WAVES) ? redS[lane] : 0.0f;
        #pragma unroll
        for (int off = 16; off > 0; off >>= 1) {
            float om = __shfl_xor(mm, off, 32);
            float os = __shfl_xor(ss, off, 32);
            float nm = fmaxf(mm, om);
            ss = ss * __expf(mm - nm) + os * __expf(om - nm);
            mm = nm;
        }
        if (lane == 0) {
            ws[(row * SPLIT + seg) * 2 + 0] = mm;
            ws[(row * SPLIT + seg) * 2 + 1] = ss;
        }
    }
}

// ---------------- Kernel B: merge SPLIT partials per row -> rowMax, logZ ----------------
__global__ __launch_bounds__(32)
void combine_kernel(float* __restrict__ ws)
{
    const int row = blockIdx.x;
    const int lane = threadIdx.x;
    float m = (lane < SPLIT) ? ws[(row * SPLIT + lane) * 2 + 0] : -INFINITY;
    float s = (lane < SPLIT) ? ws[(row * SPLIT + lane) * 2 + 1] : 0.0f;
    #pragma unroll
    for (int off = 16; off > 0; off >>= 1) {
        float om = __shfl_xor(m, off, 32);
        float os = __shfl_xor(s, off, 32);
        float nm = fmaxf(m, om);
        s = s * __expf(m - nm) + os * __expf(om - nm);
        m = nm;
    }
    if (lane == 0) {
        ws[STATS_OFF + row * 2 + 0] = m;
        ws[STATS_OFF + row * 2 + 1] = logf(s);
    }
}

// ---------------- Kernel C: write clipped logprobs (L2 reads, HBM writes, 640 WGs) ----------------
__global__ __launch_bounds__(A_THREADS)
void logprob_kernel(const float* __restrict__ logits,
                    const float* __restrict__ temps,
                    const float* __restrict__ ws,
                    float* __restrict__ out)
{
    const int row = blockIdx.x / SPLIT;
    const int seg = blockIdx.x % SPLIT;
    const int tid = threadIdx.x;
    const float invT   = 1.0f / temps[row];
    const float rowMax = ws[STATS_OFF + row * 2 + 0];
    const float logZ   = ws[STATS_OFF + row * 2 + 1];
    const float* segp  = logits + (size_t)row * VOCAB + (size_t)seg * SEG;
    float*       outp  = out + BATCH + (size_t)row * VOCAB + (size_t)seg * SEG;

    for (int i = tid * 4; i < SEG; i += A_THREADS * 4) {   // 20 iterations
        float4 v = *reinterpret_cast<const float4*>(segp + i);
        float4 o;
        o.x = fmaxf(v.x * invT - rowMax - logZ, F32_LOWEST);
        o.y = fmaxf(v.y * invT - rowMax - logZ, F32_LOWEST);
        o.z = fmaxf(v.z * invT - rowMax - logZ, F32_LOWEST);
        o.w = fmaxf(v.w * invT - rowMax - logZ, F32_LOWEST);
        *reinterpret_cast<float4*>(outp + i) = o;
    }
}

// ---------------- Kernel D: top-64 tournament + masks + Gumbel draw (L2-resident) ----------------
__global__ __launch_bounds__(D_THREADS)
void select_sample_kernel(const float* __restrict__ logits,
                          const float* __restrict__ temps,
                          const int*   __restrict__ topks,
                          const float* __restrict__ topps,
                          const float* __restrict__ minps,
                          const float* __restrict__ ws,
                          float* __restrict__ out)
{
    __shared__ float wWV[D_WAVES];
    __shared__ int   wWI[D_WAVES];
    __shared__ int   wWT[D_WAVES];
    __shared__ float topV[KMAX];
    __shared__ int   topI[KMAX];
    __shared__ float psBuf[KMAX];
    __shared__ float bSelV;
    __shared__ int   bSelI, bSelT;

    const int row  = blockIdx.x;
    const int tid  = threadIdx.x;
    const int lane = tid & 31;
    const int wave = tid >> 5;
    const float invT   = 1.0f / temps[row];
    const float rowMax = ws[STATS_OFF + row * 2 + 0];
    const float logZ   = ws[STATS_OFF + row * 2 + 1];
    const float* rowp  = logits + (size_t)row * VOCAB;

    // per-thread chunk champion
    const int cbase = tid * CHUNK;
    float cv = -INFINITY; int ci = 0x7fffffff;
    for (int j = 0; j < CHUNK; ++j) {
        int idx = cbase + j;
        float x = rowp[idx] * invT;
        if (x > cv || (x == cv && idx < ci)) { cv = x; ci = idx; }
    }
    // 64 tournament rounds
    for (int k = 0; k < KMAX; ++k) {
        float bv = cv; int bi = ci; int bt = tid;
        #pragma unroll
        for (int off = 16; off > 0; off >>= 1) {
            float ov = __shfl_xor(bv, off, 32);
            int   oi = __shfl_xor(bi, off, 32);
            int   ot = __shfl_xor(bt, off, 32);
            if (ov > bv || (ov == bv && oi < bi)) { bv = ov; bi = oi; bt = ot; }
        }
        if (lane == 0) { wWV[wave] = bv; wWI[wave] = bi; wWT[wave] = bt; }
        __syncthreads();
        if (wave == 0) {
            float fv = (lane < D_WAVES) ? wWV[lane] : -INFINITY;
            int   fi = (lane < D_WAVES) ? wWI[lane] : 0x7fffffff;
            int   ft = (lane < D_WAVES) ? wWT[lane] : -1;
            #pragma unroll
            for (int off = 16; off > 0; off >>= 1) {
                float ov = __shfl_xor(fv, off, 32);
                int   oi = __shfl_xor(fi, off, 32);
                int   ot = __shfl_xor(ft, off, 32);
                if (ov > fv || (ov == fv && oi < fi)) { fv = ov; fi = oi; ft = ot; }
            }
            if (lane == 0) { bSelV = fv; bSelI = fi; bSelT = ft; topV[k] = fv; topI[k] = fi; }
        }
        __syncthreads();
        if (tid == bSelT) {   // winner rescans its chunk (L2 hit) for its next-best
            const float sv = bSelV; const int si = bSelI;
            float nv = -INFINITY; int ni = 0x7fffffff;
            for (int j = 0; j < CHUNK; ++j) {
                int idx = cbase + j;
                float x = rowp[idx] * invT;
                bool worse = (x < sv) || (x == sv && idx > si);
                if (worse && (x > nv || (x == nv && idx < ni))) { nv = x; ni = idx; }
            }
            cv = nv; ci = ni;
        }
        __syncthreads();
    }
    // masks + Gumbel-max draw
    if (tid == 0) {
        const int   tk = topks[row];
        const float tp = topps[row];
        const float mp = minps[row];
        float cum = 0.0f;
        for (int k = 0; k < KMAX; ++k) {
            float p = __expf(topV[k] - rowMax - logZ);
            cum += p;                              // cumsum of UNfiltered sorted probs
            float q = (k >= tk) ? 0.0f : p;        // top-k
            q = (cum - q > tp) ? 0.0f : q;         // top-p (mass before token)
            psBuf[k] = q;
        }
        const float thr = psBuf[0] * mp;           // min-p vs masked max
        float best = -INFINITY; int bestk = 0;
        for (int k = 0; k < KMAX; ++k) {
            float q = (psBuf[k] < thr) ? 0.0f : psBuf[k];
            if (q > 0.0f) {
                uint32_t h = (uint32_t)topI[k] ^ ((uint32_t)row * 0x9E3779B9u) ^ (42u * 2654435761u);
                float u = uniform_hash(h);
                float g = -logf(-logf(u));         // Gumbel noise
                float sc = logf(q) + g;
                if (sc > best) { best = sc; bestk = k; }
            }
        }
        out[row] = (float)topI[bestk];
    }
}

extern "C" void kernel_launch(void* const* d_in, const int* in_sizes, int n_in,
                              void* d_out, int out_size, void* d_ws, size_t ws_size,
                              hipStream_t stream) {
    const float* logits = (const float*)d_in[0];
    const float* temps  = (const float*)d_in[1];
    const int*   topks  = (const int*)  d_in[2];
    const float* topps  = (const float*)d_in[3];
    const float* minps  = (const float*)d_in[4];
    float* out = (float*)d_out;
    float* ws  = (float*)d_ws;
    (void)in_sizes; (void)n_in; (void)out_size; (void)ws_size;

    partial_softmax_kernel<<<dim3(BATCH * SPLIT), dim3(A_THREADS), 0, stream>>>(logits, temps, ws);
    combine_kernel<<<dim3(BATCH), dim3(32), 0, stream>>>(ws);
    logprob_kernel<<<dim3(BATCH * SPLIT), dim3(A_THREADS), 0, stream>>>(logits, temps, ws, out);
    select_sample_kernel<<<dim3(BATCH), dim3(D_THREADS), 0, stream>>>(
        logits, temps, topks, topps, minps, ws, out);
}